// hybrid_10110353015423
// MI455X (gfx1250) — compile-verified
//
#include <hip/hip_runtime.h>
#include <hip/hip_bf16.h>
#include <stdint.h>

// ---------------------------------------------------------------------------
// Fused cross-entropy + softmax-gather reduction for MI455X (gfx1250).
//
// Roofline: 4096x50257 f32 = 823 MB streamed once -> ~35us @ 23.3 TB/s.
// Pure bandwidth problem; WMMA has no role (no GEMM structure). CDNA5 paths
// used: async global->LDS DMA (global_load_async_to_lds_b128 / ASYNCcnt) in a
// 4-deep per-wave ring buffer for ~240KB of outstanding DMA per WGP, plus
// wave32 shuffle reductions.
// ---------------------------------------------------------------------------

#define BLOCK        256
#define WAVES        8
#define PIPE         4                       // ring-buffer depth (stages in flight)
#define LPW          4                       // async b128 instrs per wave per stage
#define CHUNK_BYTES  512                     // one b128 instr: 32 lanes * 16B
#define WAVE_BYTES   (LPW * CHUNK_BYTES)     // 2048 B  = 512 floats per stage
#define WAVE_FLOATS  (WAVE_BYTES / 4)
#define STAGE_FLOATS (WAVES * WAVE_FLOATS)   // 4096 floats per WG stage
#define STAGE_BYTES  (STAGE_FLOATS * 4)      // 16384 B

__device__ __forceinline__ void online_upd(float& m, float& l, float x) {
  // Online logsumexp accumulator: rare rescale branch (~2.5%/elem on normal data).
  if (x > m) { l = __fmaf_rn(l, __expf(m - x), 1.0f); m = x; }
  else       { l += __expf(x - m); }
}

__device__ __forceinline__ void async_b128(uint32_t lds_addr, uint64_t gaddr) {
  // CDNA5 async DMA: each lane copies 16 bytes from the global address in vaddr to the LDS address in vdst, tracked by ASYNCcnt.
  asm volatile("global_load_async_to_lds_b128 %0, %1, off"
               :: "v"(lds_addr), "v"(gaddr) : "memory");
}

__device__ __forceinline__ void wait_async_le(int k) {
  // In-order async completion: "<= k*LPW outstanding" == everything older done.
  switch (k) {
    case 3: asm volatile("s_wait_asynccnt 0xc" ::: "memory"); break;
    case 2: asm volatile("s_wait_asynccnt 0x8" ::: "memory"); break;
    case 1: asm volatile("s_wait_asynccnt 0x4" ::: "memory"); break;
    default: asm volatile("s_wait_asynccnt 0x0" ::: "memory"); break;
  }
}

__global__ __launch_bounds__(BLOCK) void row_stats_kernel(
    const float* __restrict__ pred,
    const int*   __restrict__ y,
    const int*   __restrict__ history,
    const int*   __restrict__ nump,
    float* __restrict__ cross_part,
    float* __restrict__ mask_part,
    float* __restrict__ local_part,
    int Bdim, int V)
{
  __shared__ __align__(16) float buf[WAVES][PIPE][WAVE_FLOATS];  // 64 KB staging
  __shared__ float red_m[WAVES], red_s[WAVES];
  __shared__ float fin_m, fin_l;

  const int row  = blockIdx.x;
  const int tid  = threadIdx.x;
  const int wave = tid >> 5;     // wave32
  const int lane = tid & 31;

  const float*   rowp  = pred + (size_t)row * (size_t)V;
  const uint64_t gbase = (uint64_t)(uintptr_t)rowp;

  // AS(3) pointer value == low 32 bits of the generic LDS address.
  const uint32_t ldsw  = (uint32_t)(uintptr_t)(&buf[wave][0][0]) + (uint32_t)lane * 16u;
  const uint64_t goff  = (uint64_t)(wave * WAVE_BYTES + lane * 16);

  const int nstage = V / STAGE_FLOATS;

  float m = -3.0e38f, l = 0.0f;

  // Prologue: fill up to PIPE-1 stages of the ring.
  const int npre = (nstage < (PIPE - 1)) ? nstage : (PIPE - 1);
  for (int t = 0; t < npre; ++t) {
    const uint32_t ldst = ldsw + (uint32_t)(t * WAVE_BYTES);
    const uint64_t ga   = gbase + (uint64_t)t * STAGE_BYTES + goff;
#pragma unroll
    for (int i = 0; i < LPW; ++i)
      async_b128(ldst + (uint32_t)(i * CHUNK_BYTES), ga + (uint64_t)(i * CHUNK_BYTES));
  }

  for (int s = 0; s < nstage; ++s) {
    const int t = s + (PIPE - 1);
    if (t < nstage) {
      const uint32_t ldst = ldsw + (uint32_t)((t & (PIPE - 1)) * WAVE_BYTES);
      const uint64_t ga   = gbase + (uint64_t)t * STAGE_BYTES + goff;
#pragma unroll
      for (int i = 0; i < LPW; ++i)
        async_b128(ldst + (uint32_t)(i * CHUNK_BYTES), ga + (uint64_t)(i * CHUNK_BYTES));
    }
    // Stages still outstanding beyond stage s (clamped in the drain phase).
    int rem = nstage - 1 - s;
    if (rem > PIPE - 1) rem = PIPE - 1;
    wait_async_le(rem);

    const int par = s & (PIPE - 1);
#pragma unroll
    for (int i = 0; i < LPW; ++i) {
      const float4 v = *reinterpret_cast<const float4*>(&buf[wave][par][i * 128 + lane * 4]);
      online_upd(m, l, v.x);
      online_upd(m, l, v.y);
      online_upd(m, l, v.z);
      online_upd(m, l, v.w);
    }
  }

  // Tail (V % STAGE_FLOATS) via plain global loads.
  for (int idx = nstage * STAGE_FLOATS + tid; idx < V; idx += BLOCK)
    online_upd(m, l, rowp[idx]);

  // Wave32 (m,l) reduction via shuffles.
#pragma unroll
  for (int off = 16; off > 0; off >>= 1) {
    const float mo = __shfl_xor(m, off, 32);
    const float lo = __shfl_xor(l, off, 32);
    const float M2 = fmaxf(m, mo);
    l = l * __expf(m - M2) + lo * __expf(mo - M2);
    m = M2;
  }
  if (lane == 0) { red_m[wave] = m; red_s[wave] = l; }
  __syncthreads();
  if (tid == 0) {
    float M = red_m[0], L = red_s[0];
#pragma unroll
    for (int w = 1; w < WAVES; ++w) {
      const float M2 = fmaxf(M, red_m[w]);
      L = L * __expf(M - M2) + red_s[w] * __expf(red_m[w] - M2);
      M = M2;
    }
    fin_m = M; fin_l = L;
  }
  __syncthreads();

  const float M    = fin_m;
  const float invL = 1.0f / fin_l;
  const int   num  = *nump;

  // local = sum_n softmax(pred[row])[history[n,row]]
  float localv = 0.0f;
  for (int n = tid; n < num; n += BLOCK) {
    const int h = history[(size_t)n * (size_t)Bdim + row];
    localv += __expf(rowp[h] - M) * invL;
  }
#pragma unroll
  for (int off = 16; off > 0; off >>= 1)
    localv += __shfl_xor(localv, off, 32);
  if (lane == 0) red_s[wave] = localv;  // safe: prior read was before last barrier
  __syncthreads();

  if (tid == 0) {
    float lsum = 0.0f;
#pragma unroll
    for (int w = 0; w < WAVES; ++w) lsum += red_s[w];

    float crossv = 0.0f, maskv = 0.0f;
    const int yy = y[row];
    if (yy != 0) {
      crossv = -(rowp[yy] - M - logf(fin_l));  // -log_softmax at label
      maskv  = 1.0f;
    }
    cross_part[row] = crossv;
    mask_part[row]  = maskv;
    local_part[row] = lsum;
  }
}

__global__ __launch_bounds__(BLOCK) void finalize_kernel(
    const float* __restrict__ cross_part,
    const float* __restrict__ mask_part,
    const float* __restrict__ local_part,
    float* __restrict__ out, int Bdim)
{
  __shared__ float sc[BLOCK], sm[BLOCK], sl[BLOCK];
  float a = 0.0f, b = 0.0f, c = 0.0f;
  for (int i = threadIdx.x; i < Bdim; i += BLOCK) {
    a += cross_part[i];
    b += mask_part[i];
    c += local_part[i];
  }
  const int t = threadIdx.x;
  sc[t] = a; sm[t] = b; sl[t] = c;
  __syncthreads();
  for (int off = BLOCK / 2; off > 0; off >>= 1) {
    if (t < off) { sc[t] += sc[t + off]; sm[t] += sm[t + off]; sl[t] += sl[t + off]; }
    __syncthreads();
  }
  if (t == 0) out[0] = sc[0] / fmaxf(sm[0], 1.0f) + sl[0];
}

extern "C" void kernel_launch(void* const* d_in, const int* in_sizes, int n_in,
                              void* d_out, int out_size, void* d_ws, size_t ws_size,
                              hipStream_t stream) {
  (void)n_in; (void)out_size; (void)ws_size;
  const float* pred    = (const float*)d_in[0];
  const int*   y       = (const int*)d_in[1];
  const int*   history = (const int*)d_in[2];
  const int*   nump    = (const int*)d_in[3];

  const int Bdim = in_sizes[1];               // y has B elements
  const int V    = in_sizes[0] / Bdim;        // pred is B*V

  float* ws         = (float*)d_ws;
  float* cross_part = ws;
  float* mask_part  = ws + Bdim;
  float* local_part = ws + 2 * Bdim;

  row_stats_kernel<<<Bdim, BLOCK, 0, stream>>>(
      pred, y, history, nump, cross_part, mask_part, local_part, Bdim, V);
  finalize_kernel<<<1, BLOCK, 0, stream>>>(
      cross_part, mask_part, local_part, (float*)d_out, Bdim);
}